// ROIPooling_40389872452212
// MI455X (gfx1250) — compile-verified
//
#include <hip/hip_runtime.h>
#include <cstdint>

// ROI max-pool for MI455X (gfx1250), staged through LDS with the Tensor Data Mover.
// features: (8,128,50,50) f32, rois: (256,5) f32 [b, x1, y1, x2, y2], out: (256,128,7,7) f32.

typedef __attribute__((ext_vector_type(4))) unsigned int u32x4;
typedef __attribute__((ext_vector_type(8))) int          i32x8;
typedef __attribute__((ext_vector_type(4))) int          i32x4;

#define NB    8
#define NC    128
#define NH    50
#define NW    50
#define PH    7
#define PW    7
#define CBLK  4      // channels staged per TDM descriptor
#define NROIS 256

__global__ __launch_bounds__(32) void
roi_pool_tdm_kernel(const float* __restrict__ features,
                    const float* __restrict__ rois,
                    float* __restrict__ out) {
    // Worst-case staged tile: 4 channels x 50 x 50 f32 = 40000 B.
    __shared__ float tile[CBLK * NH * NW];

    const int n    = blockIdx.y;   // ROI index
    const int cb   = blockIdx.x;   // channel block (0..31)
    const int lane = threadIdx.x;  // wave32 lane

    // ---- ROI parameters (uniform across the wave; scalar loads) ----
    const float* r = rois + n * 5;
    const int b  = (int)r[0];                 // truncation, matches astype(int32)
    const int sw = (int)rintf(r[1]);          // jnp.round = round-half-even = rintf
    const int sh = (int)rintf(r[2]);
    const int ew = (int)rintf(r[3]);
    const int eh = (int)rintf(r[4]);

    const int   spanW = max(ew - sw + 1, 1);
    const int   spanH = max(eh - sh + 1, 1);
    const float bin_w = (float)spanW / (float)PW;   // f32 math, as in reference
    const float bin_h = (float)spanH / (float)PH;

    // Union of all bins for this ROI, clipped in-bounds (so TDM never reads OOB).
    const int c0 = min(max(sw, 0), NW);
    const int c1 = min(max(sw + spanW, 0), NW);
    const int r0 = min(max(sh, 0), NH);
    const int r1 = min(max(sh + spanH, 0), NH);
    const int Rw = max(c1 - c0, 1);
    const int Rh = max(r1 - r0, 1);

    // ---- Build Tensor DMA Descriptor (D#): 3-D tile (CBLK x Rh x Rw) of the
    //      NCHW feature tensor, strides (NH*NW, NW, 1) elements, 4 B/elem. ----
    const uint64_t gaddr = (uint64_t)(uintptr_t)features +
        4ull * ((((uint64_t)b * NC + (uint64_t)cb * CBLK) * NH + (uint64_t)r0) * NW
                + (uint64_t)c0);
    // Low 32 bits of a generic LDS pointer = workgroup-relative LDS byte address.
    const unsigned lds_addr = (unsigned)(uintptr_t)&tile[0];

    u32x4 g0;
    g0[0] = 1u;                                            // count=1 (valid), user mode
    g0[1] = lds_addr;                                      // lds_addr[31:0]
    g0[2] = (unsigned)(gaddr & 0xFFFFFFFFull);             // global_addr[31:0]
    g0[3] = (unsigned)((gaddr >> 32) & 0x01FFFFFFull)      // global_addr[56:32]
          | (2u << 30);                                    // type = 2 ("image")

    const unsigned td0 = (unsigned)Rw;        // tensor_dim0 (== tile -> no OOB)
    const unsigned td1 = (unsigned)Rh;        // tensor_dim1
    const unsigned st0 = NW;                  // tensor_dim0_stride (elements)
    const unsigned st1 = NH * NW;             // tensor_dim1_stride (elements)

    i32x8 g1;
    g1[0] = (int)(2u << 16);                                   // wg_mask=0, data_size=2 (4 B)
    g1[1] = (int)((td0 & 0xFFFFu) << 16);                      // td0[15:0] at bits[63:48]
    g1[2] = (int)((td0 >> 16) | ((td1 & 0xFFFFu) << 16));      // td0[31:16] | td1[15:0]
    g1[3] = (int)((td1 >> 16) | ((unsigned)Rw << 16));         // td1[31:16] | tile_dim0
    g1[4] = (int)((unsigned)Rh | ((unsigned)CBLK << 16));      // tile_dim1 | tile_dim2
    g1[5] = (int)st0;                                          // dim0 stride [31:0]
    g1[6] = (int)((st1 & 0xFFFFu) << 16);                      // dim0 str[47:32]=0 | dim1 str[15:0]
    g1[7] = (int)(st1 >> 16);                                  // dim1 stride [47:16]

    i32x4 g2; g2[0] = CBLK; g2[1] = 0; g2[2] = 0; g2[3] = 0;   // tensor_dim2=CBLK, rest unused
    i32x4 g3; g3[0] = 0;    g3[1] = 0; g3[2] = 0; g3[3] = 0;

    // clang-23 form carries a 5th SGPR group (ISA VADDR4, architecturally unused) -> zeros.
    i32x8 g4; g4[0] = 0; g4[1] = 0; g4[2] = 0; g4[3] = 0;
              g4[4] = 0; g4[5] = 0; g4[6] = 0; g4[7] = 0;

    // One DMA stages the whole (CBLK,Rh,Rw) window into LDS, packed contiguously.
    __builtin_amdgcn_tensor_load_to_lds(g0, g1, g2, g3, g4, 0);
    __builtin_amdgcn_s_wait_tensorcnt(0);      // TENSORcnt == 0 -> data resident in LDS
    asm volatile("" ::: "memory");             // keep LDS reads below the wait

    // ---- Max-pool 4 channels x 7 x 7 bins from LDS; coalesced stores ----
    #pragma unroll 1
    for (int i = lane; i < CBLK * PH * PW; i += 32) {
        const int cl  = i / (PH * PW);
        const int bin = i - cl * (PH * PW);
        const int ph  = bin / PW;
        const int pw  = bin - ph * PW;

        const int ws = min(max((int)floorf((float)pw * bin_w) + sw, 0), NW);
        const int we = min(max((int)ceilf((float)(pw + 1) * bin_w) + sw, 0), NW);
        const int hs = min(max((int)floorf((float)ph * bin_h) + sh, 0), NH);
        const int he = min(max((int)ceilf((float)(ph + 1) * bin_h) + sh, 0), NH);

        float v;
        if (he <= hs || we <= ws) {
            v = 0.0f;                          // empty bin rule from reference
        } else {
            v = -__builtin_inff();
            for (int h = hs; h < he; ++h) {
                const float* rowp = &tile[(cl * Rh + (h - r0)) * Rw];
                for (int w = ws; w < we; ++w) v = fmaxf(v, rowp[w - c0]);
            }
        }
        out[(((size_t)n * NC + (size_t)cb * CBLK + cl) * PH + ph) * PW + pw] = v;
    }
}

extern "C" void kernel_launch(void* const* d_in, const int* in_sizes, int n_in,
                              void* d_out, int out_size, void* d_ws, size_t ws_size,
                              hipStream_t stream) {
    (void)in_sizes; (void)n_in; (void)d_ws; (void)ws_size; (void)out_size;
    const float* features = (const float*)d_in[0];  // 8*128*50*50
    const float* rois     = (const float*)d_in[1];  // 256*5
    float*       out      = (float*)d_out;          // 256*128*7*7

    dim3 grid(NC / CBLK, NROIS);   // (32 channel blocks, 256 ROIs)
    roi_pool_tdm_kernel<<<grid, dim3(32), 0, stream>>>(features, rois, out);
}